// MILoss_46205258170959
// MI455X (gfx1250) — compile-verified
//
#include <hip/hip_runtime.h>
#include <hip/hip_bf16.h>

// ---- constants from the reference -------------------------------------------
// BIN_NUM=40, K=0.5, SIGMA=0.025
// INV      = 1/(2*pi*sigma^2) = 254.64790894703254
// MINN     = exp(-K^2)*INV    = 198.31999086396546
// 0.5/sigma^2 = 800
#define BINS      40
#define NBINS2    1600
#define INV_2PIS2 254.64790894703254f
#define MINN      198.31999086396546f
#define HALF_IS2  800.0f
#define P_BLOCKS  64     // partial histograms per batch (multiple of 4 for WMMA K)

typedef __attribute__((ext_vector_type(2))) float v2f;
typedef __attribute__((ext_vector_type(8))) float v8f;

// -----------------------------------------------------------------------------
// Kernel 1: sparse Parzen histogram. Each sample's relu(g - MINN) is exactly
// zero outside the 3x3 bin neighborhood (support radius 0.0177 < pitch 0.025),
// so evaluating 9 candidate bins reproduces the reference dense sum exactly.
// Per-block LDS histogram, written out as a partial for deterministic reduce.
// -----------------------------------------------------------------------------
__global__ void mi_hist_kernel(const float* __restrict__ x,
                               const float* __restrict__ y,
                               const int*   __restrict__ idx,
                               float* __restrict__ partial,
                               int N, int nvox, int chunk) {
    __shared__ float hist[NBINS2];
    const int tid = threadIdx.x;
    const int b   = blockIdx.y;
    const int blk = blockIdx.x;

    for (int i = tid; i < NBINS2; i += blockDim.x) hist[i] = 0.0f;
    __syncthreads();

    const int s0 = blk * chunk;
    int s1 = s0 + chunk; if (s1 > N) s1 = N;

    const float* xb = x + (size_t)b * nvox;
    const float* yb = y + (size_t)b * nvox;
    const int*   ib = idx + (size_t)b * N;

    for (int s = s0 + tid; s < s1; s += blockDim.x) {
        const int   v  = ib[s];
        const float dx = xb[v];
        const float dy = yb[v];
        const int i0 = (int)(dx * (float)BINS);
        const int j0 = (int)(dy * (float)BINS);
#pragma unroll
        for (int di = -1; di <= 1; ++di) {
            const int i = i0 + di;
            if (i < 0 || i >= BINS) continue;
            const float mx = dx - (float)(2 * i + 1) * 0.0125f;   // dx - mu1[i]
            const float mx2 = mx * mx;
#pragma unroll
            for (int dj = -1; dj <= 1; ++dj) {
                const int j = j0 + dj;
                if (j < 0 || j >= BINS) continue;
                const float my = dy - (float)(2 * j + 1) * 0.0125f; // dy - mu2[j]
                const float g  = INV_2PIS2 * __expf(-(mx2 + my * my) * HALF_IS2);
                const float c  = g - MINN;
                if (c > 0.0f) atomicAdd(&hist[i * BINS + j], c);
            }
        }
    }
    __syncthreads();

    float* dst = partial + ((size_t)(b * gridDim.x + blk)) * NBINS2;
    for (int i = tid; i < NBINS2; i += blockDim.x) dst[i] = hist[i];
}

// -----------------------------------------------------------------------------
// Kernel 2: deterministic cross-block reduction as a GEMM column-sum on the
// matrix core: D(16x16) += ones(16x4) x partial(4x16) via V_WMMA_F32_16X16X4_F32.
// One wave32 reduces 16 histogram columns over all P_BLOCKS partials (K=4/step).
// Launch: 25 blocks x 256 threads = 200 full waves = 2 batches * 100 col-groups.
// -----------------------------------------------------------------------------
__global__ void mi_reduce_wmma_kernel(const float* __restrict__ partial,
                                      float* __restrict__ hgram, int P) {
    const int lane  = threadIdx.x & 31;
    const int wid   = (blockIdx.x * blockDim.x + threadIdx.x) >> 5; // 0..199
    const int batch = wid / 100;
    const int j0    = (wid % 100) * 16;
    const int n     = lane & 15;          // B-matrix column (histogram bin)
    const int kh    = (lane >> 4) << 1;   // K rows {0,1} or {2,3} per ISA layout

    v2f a; a.x = 1.0f; a.y = 1.0f;        // ones A-matrix (16x4)
    v8f c = {};

    const float* base = partial + (size_t)batch * P * NBINS2 + j0 + n;
    for (int p = 0; p < P; p += 4) {
        v2f bv;
        bv.x = base[(size_t)(p + kh)     * NBINS2];
        bv.y = base[(size_t)(p + kh + 1) * NBINS2];
        c = __builtin_amdgcn_wmma_f32_16x16x4_f32(
                /*neg_a=*/false, a, /*neg_b=*/false, bv,
                /*c_mod=*/(short)0, c, /*reuse_a=*/false, /*reuse_b=*/false);
    }
    // Every D row holds the same column sums; VGPR0, lanes 0-15 = (M=0, N=lane).
    if (lane < 16) hgram[batch * NBINS2 + j0 + n] = c[0];
}

// -----------------------------------------------------------------------------
// Kernel 3: single-block MI evaluation: pxy = h/sum(h), marginals, entropy terms.
// -----------------------------------------------------------------------------
__global__ void mi_final_kernel(const float* __restrict__ hgram,
                                float* __restrict__ out, int B) {
    __shared__ float pxy[NBINS2];
    __shared__ float red[256];
    __shared__ float px[BINS];
    __shared__ float py[BINS];
    __shared__ float s_inv;
    const int tid = threadIdx.x;

    float total = 0.0f;
    for (int b = 0; b < B; ++b) {
        for (int i = tid; i < NBINS2; i += 256) pxy[i] = hgram[b * NBINS2 + i];
        __syncthreads();

        // total mass
        float ps = 0.0f;
        for (int i = tid; i < NBINS2; i += 256) ps += pxy[i];
        red[tid] = ps;
        __syncthreads();
        for (int o = 128; o > 0; o >>= 1) {
            if (tid < o) red[tid] += red[tid + o];
            __syncthreads();
        }
        if (tid == 0) s_inv = 1.0f / red[0];
        __syncthreads();

        const float inv = s_inv;
        for (int i = tid; i < NBINS2; i += 256) pxy[i] *= inv;
        __syncthreads();

        // marginals
        if (tid < BINS) {
            float acc = 0.0f;
            for (int j = 0; j < BINS; ++j) acc += pxy[tid * BINS + j];
            px[tid] = acc;
        } else if (tid >= 64 && tid < 64 + BINS) {
            const int j = tid - 64;
            float acc = 0.0f;
            for (int i = 0; i < BINS; ++i) acc += pxy[i * BINS + j];
            py[j] = acc;
        }
        __syncthreads();

        // -sum(pxy*log(pxy+eps) - pxy*log(px*py+eps))
        float pl = 0.0f;
        for (int k = tid; k < NBINS2; k += 256) {
            const float p  = pxy[k];
            const float pp = px[k / BINS] * py[k % BINS];
            pl += p * __logf(p + 1e-9f) - p * __logf(pp + 1e-9f);
        }
        red[tid] = pl;
        __syncthreads();
        for (int o = 128; o > 0; o >>= 1) {
            if (tid < o) red[tid] += red[tid + o];
            __syncthreads();
        }
        if (tid == 0) total += -red[0];
        __syncthreads();
    }
    if (tid == 0) out[0] = total / (float)B;
}

// -----------------------------------------------------------------------------
extern "C" void kernel_launch(void* const* d_in, const int* in_sizes, int n_in,
                              void* d_out, int out_size, void* d_ws, size_t ws_size,
                              hipStream_t stream) {
    const float* fix = (const float*)d_in[0];
    const float* reg = (const float*)d_in[1];
    const int*   idx = (const int*)d_in[2];
    float*       out = (float*)d_out;

    const int B    = 2;
    const int nvox = in_sizes[0] / B;   // 1,048,576
    const int N    = in_sizes[2] / B;   // 200,000
    const int P    = P_BLOCKS;

    float* partial = (float*)d_ws;                       // [B * P * 1600]
    float* hgram   = partial + (size_t)B * P * NBINS2;   // [B * 1600]

    const int chunk = (N + P - 1) / P;

    dim3 grid1(P, B);
    mi_hist_kernel<<<grid1, 256, 0, stream>>>(fix, reg, idx, partial, N, nvox, chunk);

    // 200 waves total: B(2) * 100 column-groups of 16 bins
    mi_reduce_wmma_kernel<<<25, 256, 0, stream>>>(partial, hgram, P);

    mi_final_kernel<<<1, 256, 0, stream>>>(hgram, out, B);
}